// Q_GCNLayer_68247030333446
// MI455X (gfx1250) — compile-verified
//
#include <hip/hip_runtime.h>
#include <hip/hip_bf16.h>

typedef __attribute__((ext_vector_type(2))) float v2f;
typedef __attribute__((ext_vector_type(8))) float v8f;

#define IN_DIM  128
#define OUT_DIM 128

// ---------------------------------------------------------------------------
// Kernel 1: zero the output accumulator (float4-wide).
// ---------------------------------------------------------------------------
__global__ void zero_f32_kernel(float4* __restrict__ p, int n4) {
    int i = blockIdx.x * blockDim.x + threadIdx.x;
    if (i < n4) p[i] = make_float4(0.f, 0.f, 0.f, 0.f);
}

// ---------------------------------------------------------------------------
// Kernel 2: h = x @ W + b  via V_WMMA_F32_16X16X4_F32 (full fp32 precision).
// One wave computes one 16x16 tile of h. 8 waves/block cover the 8 N-tiles of
// the same 16 rows (A-operand reuse through L0/L1).
//
// Per ISA 7.12.2 (wave32):
//   A 16x4 f32: lane<16 -> M=lane, {K=k0,k0+1}; lane>=16 -> M=lane-16, {K=k0+2,k0+3}
//   B 4x16 f32: same striping over N
//   C/D 16x16:  VGPR i -> row (half*8+i), col (lane&15)
// ---------------------------------------------------------------------------
__global__ void gemm_bias_wmma_kernel(const float* __restrict__ x,
                                      const float* __restrict__ W,
                                      const float* __restrict__ bias,
                                      float* __restrict__ h,
                                      int n_rows) {
    const int wave   = threadIdx.x >> 5;
    const int lane   = threadIdx.x & 31;
    const int tile   = blockIdx.x * (blockDim.x >> 5) + wave;
    const int m_tile = tile >> 3;                // OUT_DIM/16 == 8 N-tiles
    const int n_tile = tile & 7;
    const int rowBase = m_tile * 16;
    if (rowBase >= n_rows) return;
    const int colBase = n_tile * 16;
    const int half = lane >> 4;                  // 0: K pair {0,1}, 1: {2,3}
    const int lr   = lane & 15;

    const float* __restrict__ xrow = x + (long)(rowBase + lr) * IN_DIM;
    const float* __restrict__ wcol = W + (colBase + lr);

    v8f acc = {};
#pragma unroll
    for (int k0 = 0; k0 < IN_DIM; k0 += 4) {
        const int ka = k0 + half * 2;
        v2f a;                                   // contiguous -> global_load_b64
        a.x = xrow[ka];
        a.y = xrow[ka + 1];
        v2f bfr;
        bfr.x = wcol[(long)ka * OUT_DIM];
        bfr.y = wcol[(long)(ka + 1) * OUT_DIM];
        acc = __builtin_amdgcn_wmma_f32_16x16x4_f32(
            /*neg_a=*/false, a, /*neg_b=*/false, bfr,
            /*c_mod=*/(short)0, acc, /*reuse_a=*/false, /*reuse_b=*/false);
    }

    const float bv = bias[colBase + lr];
#pragma unroll
    for (int i = 0; i < 8; ++i) {
        const int m = rowBase + half * 8 + i;
        h[(long)m * OUT_DIM + colBase + lr] = acc[i] + bv;
    }
}

// ---------------------------------------------------------------------------
// Kernel 3: edge scatter. One wave per edge: lane L gathers float4 at
// h[cols[e]*128 + 4L] (one coalesced 512B wave transaction, L2-resident since
// h = 51 MB < 192 MB L2), scales by vals[e], and HW-atomic-adds into
// out[rows[e]] (also L2-resident).
// ---------------------------------------------------------------------------
__global__ void scatter_edges_kernel(const float* __restrict__ h,
                                     const float* __restrict__ vals,
                                     const int*   __restrict__ rows,
                                     const int*   __restrict__ cols,
                                     float* __restrict__ out,
                                     int n_edges) {
    const long tid = (long)blockIdx.x * blockDim.x + threadIdx.x;
    const int  e   = (int)(tid >> 5);            // 32 lanes per edge
    if (e >= n_edges) return;
    const int lane = (int)(tid & 31);

    const int   r = rows[e];
    const int   c = cols[e];
    const float v = vals[e];

    const float4 hv = ((const float4*)(h + (long)c * OUT_DIM))[lane];
    float* __restrict__ op = out + (long)r * OUT_DIM + lane * 4;

    unsafeAtomicAdd(op + 0, v * hv.x);           // global_atomic_add_f32
    unsafeAtomicAdd(op + 1, v * hv.y);
    unsafeAtomicAdd(op + 2, v * hv.z);
    unsafeAtomicAdd(op + 3, v * hv.w);
}

// ---------------------------------------------------------------------------
// Launch: zero(out) -> GEMM(x,W,b -> h in d_ws) -> scatter(edges -> out)
// ---------------------------------------------------------------------------
extern "C" void kernel_launch(void* const* d_in, const int* in_sizes, int n_in,
                              void* d_out, int out_size, void* d_ws, size_t ws_size,
                              hipStream_t stream) {
    const float* x    = (const float*)d_in[0];
    const float* W    = (const float*)d_in[1];
    const float* bias = (const float*)d_in[2];
    const float* vals = (const float*)d_in[3];
    const int*   rows = (const int*)d_in[4];
    const int*   cols = (const int*)d_in[5];
    float*       out  = (float*)d_out;
    float*       h    = (float*)d_ws;            // 100000*128*4 = 51.2 MB

    const int n_nodes = in_sizes[0] / IN_DIM;    // 100000
    const int n_edges = in_sizes[3];             // 1600000

    // 1) zero out accumulator
    const int n4 = out_size / 4;
    zero_f32_kernel<<<(n4 + 255) / 256, 256, 0, stream>>>((float4*)out, n4);

    // 2) dense h = x@W + b (fp32 WMMA)
    const int m_tiles = (n_nodes + 15) / 16;
    const int tiles   = m_tiles * (OUT_DIM / 16);
    const int waves_per_block = 8;               // 256 threads
    const int blocks = (tiles + waves_per_block - 1) / waves_per_block;
    gemm_bias_wmma_kernel<<<blocks, waves_per_block * 32, 0, stream>>>(
        x, W, bias, h, n_nodes);

    // 3) sparse scatter-add over edges
    const long sthreads = (long)n_edges * 32;
    scatter_edges_kernel<<<(int)((sthreads + 255) / 256), 256, 0, stream>>>(
        h, vals, rows, cols, out, n_edges);
}